// Graphsage_86260123173600
// MI455X (gfx1250) — compile-verified
//
#include <hip/hip_runtime.h>
#include <cstdint>

typedef __attribute__((ext_vector_type(2))) float v2f;
typedef __attribute__((ext_vector_type(8))) float v8f;

constexpr int F_INC = 256;   // input features
constexpr int HDIM  = 128;   // hidden
constexpr int CDIM  = 40;    // classes
constexpr int CPAD  = 48;    // padded to 3 x 16 WMMA col tiles

__device__ __forceinline__ v8f splat8(float x) {
  v8f r = {x, x, x, x, x, x, x, x};
  return r;
}

// ---------------- graph build: histogram -> scan -> bucket (CSR) --------------

__global__ void hist_kernel(const int* __restrict__ rows, int* __restrict__ deg, int E) {
  int e = blockIdx.x * blockDim.x + threadIdx.x;
  if (e < E) atomicAdd(&deg[rows[e]], 1);
}

__global__ void scan_block_sums(const int* __restrict__ deg, int* __restrict__ bsums, int N) {
  __shared__ int sm[256];
  int tid = threadIdx.x;
  int base = blockIdx.x * 1024 + tid * 4;
  int s = 0;
#pragma unroll
  for (int j = 0; j < 4; ++j) { int i = base + j; if (i < N) s += deg[i]; }
  sm[tid] = s;
  __syncthreads();
  for (int off = 128; off > 0; off >>= 1) {
    if (tid < off) sm[tid] += sm[tid + off];
    __syncthreads();
  }
  if (tid == 0) bsums[blockIdx.x] = sm[0];
}

__global__ void scan_offsets(const int* __restrict__ bsums, int* __restrict__ boff, int nb) {
  if (blockIdx.x == 0 && threadIdx.x == 0) {
    int run = 0;
    for (int i = 0; i < nb; ++i) { boff[i] = run; run += bsums[i]; }
  }
}

__global__ void scan_final(const int* __restrict__ deg, const int* __restrict__ boff,
                           int* __restrict__ row_start, float* __restrict__ dinv, int N) {
  __shared__ int sm[256];
  int tid = threadIdx.x;
  int base = blockIdx.x * 1024 + tid * 4;
  int v[4]; int s = 0;
#pragma unroll
  for (int j = 0; j < 4; ++j) { int i = base + j; v[j] = (i < N) ? deg[i] : 0; s += v[j]; }
  sm[tid] = s;
  __syncthreads();
  for (int off = 1; off < 256; off <<= 1) {   // Hillis-Steele inclusive scan
    int t = (tid >= off) ? sm[tid - off] : 0;
    __syncthreads();
    sm[tid] += t;
    __syncthreads();
  }
  int excl = sm[tid] - s + boff[blockIdx.x];
#pragma unroll
  for (int j = 0; j < 4; ++j) {
    int i = base + j;
    if (i < N) {
      row_start[i] = excl;
      dinv[i] = (v[j] > 0) ? (1.0f / (float)v[j]) : 0.0f;
      excl += v[j];
    }
  }
}

__global__ void copy_i32(const int* __restrict__ src, int* __restrict__ dst, int n) {
  int i = blockIdx.x * blockDim.x + threadIdx.x;
  if (i < n) dst[i] = src[i];
}

__global__ void bucket_kernel(const int* __restrict__ rows, const int* __restrict__ cols,
                              int* __restrict__ cursor, int* __restrict__ col_sorted, int E) {
  int e = blockIdx.x * blockDim.x + threadIdx.x;
  if (e < E) {
    int slot = atomicAdd(&cursor[rows[e]], 1);
    col_sorted[slot] = cols[e];
  }
}

// ---------------- GEMM1: h1[N,128] = x[N,256] @ W1[256,128] + b1 (WMMA f32) ----

// 256 threads (8 waves). Block tile = 32 rows x 128 cols. Wave w: col tile 16*w,
// two row sub-tiles (B fragment reused across both). x tile staged in LDS.
__global__ void gemm1_kernel(const float* __restrict__ x, const float* __restrict__ W1,
                             const float* __restrict__ b1, float* __restrict__ h1, int N) {
  __shared__ float lx[32 * 260];              // stride 260: 16B-aligned rows, bank-clean
  const int tid = threadIdx.x;
  const int r0 = blockIdx.x * 32;
  const bool full = (r0 + 32 <= N);

  if (full) {
    for (int i = tid * 4; i < 32 * F_INC; i += 256 * 4) {
      int r = i >> 8, c = i & 255;
      *(float4*)(&lx[r * 260 + c]) = *(const float4*)(x + (size_t)(r0 + r) * F_INC + c);
    }
  } else {
    for (int i = tid * 4; i < 32 * F_INC; i += 256 * 4) {
      int r = i >> 8, c = i & 255;
      int gr = min(r0 + r, N - 1);
      *(float4*)(&lx[r * 260 + c]) = *(const float4*)(x + (size_t)gr * F_INC + c);
    }
  }
  __syncthreads();

  const int wave = tid >> 5, lane = tid & 31;
  const int half = lane >> 4, m = lane & 15;
  const int col = wave * 16 + m;

  v8f acc0 = splat8(b1[col]);                 // fold bias into C
  v8f acc1 = acc0;

  for (int k0 = 0; k0 < F_INC; k0 += 4) {
    int ke = k0 + 2 * half;                   // ISA A layout: lanes0-15 K=0,1; lanes16-31 K=2,3
    v2f b;
    b.x = W1[(size_t)ke * HDIM + col];
    b.y = W1[(size_t)(ke + 1) * HDIM + col];
    v2f a0 = *(const v2f*)(&lx[m * 260 + ke]);
    v2f a1 = *(const v2f*)(&lx[(m + 16) * 260 + ke]);
    acc0 = __builtin_amdgcn_wmma_f32_16x16x4_f32(false, a0, false, b, (short)0, acc0, false, false);
    acc1 = __builtin_amdgcn_wmma_f32_16x16x4_f32(false, a1, false, b, (short)0, acc1, false, false);
  }

  // D layout: VGPR j -> row j + 8*half, col = m
  if (full) {
#pragma unroll
    for (int j = 0; j < 8; ++j) {
      int rr = r0 + j + 8 * half;
      h1[(size_t)rr * HDIM + col] = acc0[j];
      h1[(size_t)(rr + 16) * HDIM + col] = acc1[j];
    }
  } else {
#pragma unroll
    for (int j = 0; j < 8; ++j) {
      int rr0 = r0 + j + 8 * half;
      int rr1 = rr0 + 16;
      if (rr0 < N) h1[(size_t)rr0 * HDIM + col] = acc0[j];
      if (rr1 < N) h1[(size_t)rr1 * HDIM + col] = acc1[j];
    }
  }
}

// ---------------- gather1: agg1[i] = relu(dinv[i] * sum_j h1[col_j])  ----------

__global__ void gather1_kernel(const float* __restrict__ h1, const int* __restrict__ row_start,
                               const int* __restrict__ deg, const float* __restrict__ dinv,
                               const int* __restrict__ col_sorted, float* __restrict__ agg1, int N) {
  int wave = threadIdx.x >> 5, lane = threadIdx.x & 31;
  int node = blockIdx.x * 8 + wave;
  if (node >= N) return;
  int start = row_start[node];
  int cnt = deg[node];
  float4 acc = make_float4(0.f, 0.f, 0.f, 0.f);
  for (int eb = 0; eb < cnt; eb += 32) {
    int myc = (eb + lane < cnt) ? col_sorted[start + eb + lane] : 0;
    int n = min(32, cnt - eb);
    for (int j = 0; j < n; ++j) {
      // uniform broadcast -> SGPR col -> scalar address math + saddr loads
      int col = __builtin_amdgcn_readlane(myc, j);
      const float4 v = *(const float4*)(h1 + (size_t)col * HDIM + lane * 4);
      acc.x += v.x; acc.y += v.y; acc.z += v.z; acc.w += v.w;
    }
  }
  float d = dinv[node];
  float4 o;
  o.x = fmaxf(acc.x * d, 0.f); o.y = fmaxf(acc.y * d, 0.f);
  o.z = fmaxf(acc.z * d, 0.f); o.w = fmaxf(acc.w * d, 0.f);
  *(float4*)(agg1 + (size_t)node * HDIM + lane * 4) = o;
}

// ---------------- GEMM2: h2[N,40] = agg1[N,128] @ W2[128,40] + b2 (WMMA f32) ---

// 384 threads (12 waves = 4 row-tiles x 3 col-tiles). Block tile = 64 rows x 48 cols.
__global__ void gemm2_kernel(const float* __restrict__ agg1, const float* __restrict__ W2,
                             const float* __restrict__ b2, float* __restrict__ h2, int N) {
  __shared__ float la[64 * 132];              // 33.8 KB
  __shared__ float lw[HDIM * CPAD];           // 24 KB, zero-padded cols 40..47
  const int tid = threadIdx.x;
  const int r0 = blockIdx.x * 64;
  const bool full = (r0 + 64 <= N);

  for (int i = tid; i < HDIM * CPAD; i += 384) {
    int k = i / CPAD, c = i - k * CPAD;
    lw[i] = (c < CDIM) ? W2[k * CDIM + c] : 0.f;
  }
  if (full) {
    for (int i = tid * 4; i < 64 * HDIM; i += 384 * 4) {
      int r = i >> 7, c = i & 127;
      *(float4*)(&la[r * 132 + c]) = *(const float4*)(agg1 + (size_t)(r0 + r) * HDIM + c);
    }
  } else {
    for (int i = tid * 4; i < 64 * HDIM; i += 384 * 4) {
      int r = i >> 7, c = i & 127;
      int gr = r0 + r;
      float4 v = (gr < N) ? *(const float4*)(agg1 + (size_t)gr * HDIM + c)
                          : make_float4(0.f, 0.f, 0.f, 0.f);
      *(float4*)(&la[r * 132 + c]) = v;
    }
  }
  __syncthreads();

  const int wave = tid >> 5, lane = tid & 31;
  const int half = lane >> 4, m = lane & 15;
  const int rt = wave / 3;                    // row tile 0..3
  const int ct = wave - rt * 3;               // col tile 0..2
  const int col = ct * 16 + m;

  v8f acc = splat8((col < CDIM) ? b2[col] : 0.f);

  for (int k0 = 0; k0 < HDIM; k0 += 4) {
    int ke = k0 + 2 * half;
    v2f a = *(const v2f*)(&la[(rt * 16 + m) * 132 + ke]);
    v2f b;
    b.x = lw[ke * CPAD + col];
    b.y = lw[(ke + 1) * CPAD + col];
    acc = __builtin_amdgcn_wmma_f32_16x16x4_f32(false, a, false, b, (short)0, acc, false, false);
  }

  if (col < CDIM) {
    if (full) {
#pragma unroll
      for (int j = 0; j < 8; ++j) {
        int row = r0 + rt * 16 + j + 8 * half;
        h2[(size_t)row * CDIM + col] = acc[j];
      }
    } else {
#pragma unroll
      for (int j = 0; j < 8; ++j) {
        int row = r0 + rt * 16 + j + 8 * half;
        if (row < N) h2[(size_t)row * CDIM + col] = acc[j];
      }
    }
  }
}

// ---------------- gather2: agg2[i] = dinv[i] * sum_j h2[col_j] -----------------

__global__ void gather2_kernel(const float* __restrict__ h2, const int* __restrict__ row_start,
                               const int* __restrict__ deg, const float* __restrict__ dinv,
                               const int* __restrict__ col_sorted, float* __restrict__ agg2, int N) {
  int wave = threadIdx.x >> 5, lane = threadIdx.x & 31;
  int node = blockIdx.x * 8 + wave;
  if (node >= N) return;
  int start = row_start[node];
  int cnt = deg[node];
  float a0 = 0.f, a1 = 0.f;                   // lane covers f=lane and (lane<8) f=lane+32
  for (int eb = 0; eb < cnt; eb += 32) {
    int myc = (eb + lane < cnt) ? col_sorted[start + eb + lane] : 0;
    int n = min(32, cnt - eb);
    for (int j = 0; j < n; ++j) {
      int col = __builtin_amdgcn_readlane(myc, j);
      const float* p = h2 + (size_t)col * CDIM;
      a0 += p[lane];
      if (lane < 8) a1 += p[lane + 32];
    }
  }
  float d = dinv[node];
  agg2[(size_t)node * CDIM + lane] = a0 * d;
  if (lane < 8) agg2[(size_t)node * CDIM + lane + 32] = a1 * d;
}

// ---------------- row-wise log_softmax -----------------------------------------

__global__ void logsoftmax_kernel(const float* __restrict__ agg2, float* __restrict__ out, int N) {
  int i = blockIdx.x * blockDim.x + threadIdx.x;
  if (i >= N) return;
  const float* p = agg2 + (size_t)i * CDIM;
  float v[CDIM];
  float mx = -3.402823466e38f;
#pragma unroll
  for (int f = 0; f < CDIM; ++f) { v[f] = p[f]; mx = fmaxf(mx, v[f]); }
  float s = 0.f;
#pragma unroll
  for (int f = 0; f < CDIM; ++f) s += __expf(v[f] - mx);
  float ls = mx + __logf(s);
  float* q = out + (size_t)i * CDIM;
#pragma unroll
  for (int f = 0; f < CDIM; ++f) q[f] = v[f] - ls;
}

// ---------------- launch --------------------------------------------------------

extern "C" void kernel_launch(void* const* d_in, const int* in_sizes, int n_in,
                              void* d_out, int out_size, void* d_ws, size_t ws_size,
                              hipStream_t stream) {
  const float* x  = (const float*)d_in[0];
  const int* rows = (const int*)d_in[1];
  const float* W1 = (const float*)d_in[2];
  const float* b1 = (const float*)d_in[3];
  const float* W2 = (const float*)d_in[4];
  const float* b2 = (const float*)d_in[5];
  float* out = (float*)d_out;

  const int H   = in_sizes[3];            // 128
  const int FIN = in_sizes[2] / H;        // 256
  const int N   = in_sizes[0] / FIN;      // 100000
  const int E   = in_sizes[1] / 2;        // 1700000 (edges + self loops)
  const int* cols = rows + E;
  (void)n_in; (void)out_size; (void)ws_size;

  // workspace carving (256B aligned)
  char* w = (char*)d_ws;
  auto carve = [&](size_t bytes) -> char* {
    char* p = w;
    w += (bytes + 255) & ~(size_t)255;
    return p;
  };
  float* h1        = (float*)carve((size_t)N * HDIM * sizeof(float));
  float* agg1      = (float*)carve((size_t)N * HDIM * sizeof(float));
  float* h2        = (float*)carve((size_t)N * CDIM * sizeof(float));
  float* agg2      = (float*)carve((size_t)N * CDIM * sizeof(float));
  float* dinv      = (float*)carve((size_t)N * sizeof(float));
  int*   deg       = (int*)carve((size_t)N * sizeof(int));
  int*   row_start = (int*)carve((size_t)N * sizeof(int));
  int*   cursor    = (int*)carve((size_t)N * sizeof(int));
  int*   col_sort  = (int*)carve((size_t)E * sizeof(int));
  const int nb = (N + 1023) / 1024;
  int*   bsums     = (int*)carve((size_t)nb * sizeof(int));
  int*   boff      = (int*)carve((size_t)nb * sizeof(int));

  // CSR build (int atomics only; reused by both layers)
  hipMemsetAsync(deg, 0, (size_t)N * sizeof(int), stream);
  hist_kernel<<<(E + 255) / 256, 256, 0, stream>>>(rows, deg, E);
  scan_block_sums<<<nb, 256, 0, stream>>>(deg, bsums, N);
  scan_offsets<<<1, 32, 0, stream>>>(bsums, boff, nb);
  scan_final<<<nb, 256, 0, stream>>>(deg, boff, row_start, dinv, N);
  copy_i32<<<(N + 255) / 256, 256, 0, stream>>>(row_start, cursor, N);
  bucket_kernel<<<(E + 255) / 256, 256, 0, stream>>>(rows, cols, cursor, col_sort, E);

  // layer 1
  gemm1_kernel<<<(N + 31) / 32, 256, 0, stream>>>(x, W1, b1, h1, N);
  gather1_kernel<<<(N + 7) / 8, 256, 0, stream>>>(h1, row_start, deg, dinv, col_sort, agg1, N);
  // layer 2
  gemm2_kernel<<<(N + 63) / 64, 384, 0, stream>>>(agg1, W2, b2, h2, N);
  gather2_kernel<<<(N + 7) / 8, 256, 0, stream>>>(h2, row_start, deg, dinv, col_sort, agg2, N);
  // output
  logsoftmax_kernel<<<(N + 255) / 256, 256, 0, stream>>>(agg2, out, N);
}